// ParametricPart_27779848471269
// MI455X (gfx1250) — compile-verified
//
#include <hip/hip_runtime.h>

// ---------------------------------------------------------------------------
// Periodic 8^3 stencil RHS via WMMA: every 1D periodic operator (central diff,
// second diff) is an 8x8 circulant; pack two 8-rows per WMMA row and multiply
// by blockdiag(D,D) so one 16x16x16 f32 WMMA product handles 32 rows.
// One wave = one batch; full tile reuse in LDS.
// Data motion uses CDNA5 async HBM<->LDS ops (ASYNCcnt): one streaming read
// and one streaming write of HBM per element, no VGPR round-trip.
// ---------------------------------------------------------------------------

#define GDIM 8
#define NPTS 512            // 8^3
#define NFIELD 1536         // 3 * 512
#define WAVES_PER_WG 8
#define BATCH_LDS 4096      // floats: u(1536) + pscal(512) + div(512) + out(1536)
#define P_LDS 1536
#define SMEM_FLOATS (P_LDS + WAVES_PER_WG * BATCH_LDS)

#define INV2DX  0.0625f     // 1/(2*8)
#define INVDX2  0.015625f   // 1/64

typedef __attribute__((ext_vector_type(2))) float v2f;
typedef __attribute__((ext_vector_type(8))) float v8f;

// Linear index of point (rowsel, col) where `col` runs along `AXIS` and
// rowsel enumerates the other two coordinates. Same bijection is used for
// gathering A fragments and scattering C fragments.
template <int AXIS>
__device__ __forceinline__ int map_idx(int rowsel, int col) {
    if (AXIS == 0) return col * 64 + rowsel;                                   // col = h
    if (AXIS == 1) return ((rowsel >> 3) << 6) + (col << 3) + (rowsel & 7);    // col = w
    return (rowsel << 3) + col;                                                // col = l
}

template <int AXIS>
__device__ __forceinline__ int a_idx(int t, int m, int k) {
    int rowsel = t * 32 + ((k & 8) ? 16 : 0) + m;
    return map_idx<AXIS>(rowsel, k & 7);
}

// 8x8 circulant operators
__device__ __forceinline__ float dmat_central(int k, int n) {
    float v = 0.f;
    if (k == ((n + 1) & 7)) v += INV2DX;
    if (k == ((n + 7) & 7)) v -= INV2DX;
    return v;
}
__device__ __forceinline__ float dmat_lap(int k, int n) {
    float v = 0.f;
    if (k == ((n + 1) & 7)) v += INVDX2;
    if (k == ((n + 7) & 7)) v += INVDX2;
    if (k == n)             v -= 2.0f * INVDX2;
    return v;
}
__device__ __forceinline__ float bdiag(int k, int n, int lap) {
    if ((k & 8) != (n & 8)) return 0.f;
    return lap ? dmat_lap(k & 7, n & 7) : dmat_central(k & 7, n & 7);
}

struct AFrag { v2f a[4]; };

template <int AXIS>
__device__ __forceinline__ AFrag load_afrag(const float* __restrict__ F, int t, int m16, int half) {
    AFrag A;
#pragma unroll
    for (int c = 0; c < 4; ++c) {
        int k0 = 4 * c + 2 * half;
        A.a[c].x = F[a_idx<AXIS>(t, m16, k0)];
        A.a[c].y = F[a_idx<AXIS>(t, m16, k0 + 1)];
    }
    return A;
}

__device__ __forceinline__ v8f wmma_chain(const AFrag& A, const v2f B[4], v8f C) {
#pragma unroll
    for (int c = 0; c < 4; ++c)
        C = __builtin_amdgcn_wmma_f32_16x16x4_f32(false, A.a[c], false, B[c],
                                                  (short)0, C, false, false);
    return C;
}

template <int AXIS>
__device__ __forceinline__ void compute_cix(int t, int m16, int half, int cix[8]) {
#pragma unroll
    for (int v = 0; v < 8; ++v) {
        int M = v + 8 * half;
        int rowsel = t * 32 + ((m16 & 8) ? 16 : 0) + M;
        cix[v] = map_idx<AXIS>(rowsel, m16 & 7);
    }
}

// one derivative-axis pass: grad_u (for u.grad_u, div), laplacian part, grad p_scal
template <int AXIS>
__device__ __forceinline__ void axis_pass(float* su, float* sp, float* sdv, float* so,
                                          const float d1[3], float d2a,
                                          int m16, int half,
                                          const v2f Bc[4], const v2f Bl[4]) {
#pragma unroll
    for (int t = 0; t < 2; ++t) {
        int cix[8];
        compute_cix<AXIS>(t, m16, half, cix);
        v8f acc = {};
#pragma unroll
        for (int i = 0; i < 3; ++i) {
            AFrag A = load_afrag<AXIS>(su + i * NPTS, t, m16, half);
            v8f C  = {};   // first derivative
            v8f C2 = {};   // second derivative (laplacian part)
            C  = wmma_chain(A, Bc, C);
            C2 = wmma_chain(A, Bl, C2);
#pragma unroll
            for (int v = 0; v < 8; ++v) {
                so[i * NPTS + cix[v]] += d1[i] * C2[v];           // D1 * lap
                acc[v] += su[i * NPTS + cix[v]] * C[v];            // u_i * d_AXIS u_i
                if (i == AXIS) sdv[cix[v]] += C[v];                // divergence
            }
        }
        // p_scal derivative along AXIS
        AFrag Ap = load_afrag<AXIS>(sp, t, m16, half);
        v8f Cp = {};
        Cp = wmma_chain(Ap, Bc, Cp);
#pragma unroll
        for (int v = 0; v < 8; ++v)
            so[AXIS * NPTS + cix[v]] += -0.001f * acc[v] - d2a * Cp[v];
    }
}

template <int AXIS>
__device__ __forceinline__ void graddiv_pass(float* sdv, float* so, float d1a,
                                             int m16, int half, const v2f Bc[4]) {
#pragma unroll
    for (int t = 0; t < 2; ++t) {
        int cix[8];
        compute_cix<AXIS>(t, m16, half, cix);
        AFrag Ad = load_afrag<AXIS>(sdv, t, m16, half);
        v8f Cd = {};
        Cd = wmma_chain(Ad, Bc, Cd);
#pragma unroll
        for (int v = 0; v < 8; ++v)
            so[AXIS * NPTS + cix[v]] += (-d1a * (1.0f / 3.0f)) * Cd[v];
    }
}

// CDNA5 async HBM -> LDS (16B per lane per issue), tracked by ASYNCcnt
__device__ __forceinline__ void async_load_b128(unsigned lds_off, unsigned long long gaddr) {
    asm volatile("global_load_async_to_lds_b128 %0, %1, off"
                 :: "v"(lds_off), "v"(gaddr) : "memory");
}
// CDNA5 async LDS -> HBM
__device__ __forceinline__ void async_store_b128(unsigned long long gaddr, unsigned lds_off) {
    asm volatile("global_store_async_from_lds_b128 %0, %1, off"
                 :: "v"(gaddr), "v"(lds_off) : "memory");
}
__device__ __forceinline__ void wait_async0() {
    asm volatile("s_wait_asynccnt 0x0" ::: "memory");
}
__device__ __forceinline__ void wait_ds0() {
    asm volatile("s_wait_dscnt 0x0" ::: "memory");
}

__global__ void __launch_bounds__(WAVES_PER_WG * 32, 1)
fluid_rhs_kernel(const float* __restrict__ state, const float* __restrict__ D1g,
                 const float* __restrict__ D2g, const float* __restrict__ pg,
                 float* __restrict__ outg, int nbatch) {
    extern __shared__ float smem[];
    const int tid  = threadIdx.x;
    const int lane = tid & 31;
    const int wave = tid >> 5;
    const int batch = blockIdx.x * WAVES_PER_WG + wave;
    const bool active = (batch < nbatch);

    // ---- async stage p field (block-shared, 384 x 16B chunks) ----
    {
        unsigned p_lds = (unsigned)(uintptr_t)smem;
        unsigned long long p_g = (unsigned long long)(uintptr_t)pg;
        for (int i = tid; i < P_LDS / 4; i += WAVES_PER_WG * 32)
            async_load_b128(p_lds + (unsigned)i * 16u, p_g + (unsigned long long)i * 16ull);
    }

    float* su  = smem + P_LDS + wave * BATCH_LDS;  // state tile   (1536)
    float* sp  = su + NFIELD;                      // p_scal       (512)
    float* sdv = sp + NPTS;                        // divergence   (512)
    float* so  = sdv + NPTS;                       // output accum (1536)

    if (active) {
        // ---- async stage state tile: 12 x b128 per lane ----
        unsigned su_lds = (unsigned)(uintptr_t)su;
        unsigned long long gu = (unsigned long long)(uintptr_t)(state + (size_t)batch * NFIELD);
#pragma unroll
        for (int i = 0; i < 12; ++i) {
            unsigned off = (unsigned)(lane + i * 32) * 16u;
            async_load_b128(su_lds + off, gu + off);
        }
        // zero div + out (contiguous 2048 floats) while loads are in flight
        const float4 z4 = make_float4(0.f, 0.f, 0.f, 0.f);
        for (int i = lane; i < (NPTS + NFIELD) / 4; i += 32)
            ((float4*)sdv)[i] = z4;
    }

    wait_async0();          // state + p resident in LDS
    __syncthreads();        // p visible to all waves

    if (active) {
        const float d1[3] = {D1g[0], D1g[1], D1g[2]};
        const float d2[3] = {D2g[0], D2g[1], D2g[2]};

        // p_scal = sum_i u_i * p_i
        for (int pt = lane; pt < NPTS; pt += 32)
            sp[pt] = su[pt] * smem[pt] + su[NPTS + pt] * smem[NPTS + pt]
                   + su[2 * NPTS + pt] * smem[2 * NPTS + pt];

        const int half = lane >> 4;
        const int m16  = lane & 15;

        // constant operator fragments: blockdiag(D,D) for central & second diff
        v2f Bc[4], Bl[4];
#pragma unroll
        for (int c = 0; c < 4; ++c) {
            int kb = 4 * c + 2 * half;
            Bc[c].x = bdiag(kb,     m16, 0);
            Bc[c].y = bdiag(kb + 1, m16, 0);
            Bl[c].x = bdiag(kb,     m16, 1);
            Bl[c].y = bdiag(kb + 1, m16, 1);
        }

        asm volatile("" ::: "memory");

        axis_pass<0>(su, sp, sdv, so, d1, d2[0], m16, half, Bc, Bl);
        axis_pass<1>(su, sp, sdv, so, d1, d2[1], m16, half, Bc, Bl);
        axis_pass<2>(su, sp, sdv, so, d1, d2[2], m16, half, Bc, Bl);

        asm volatile("" ::: "memory");

        graddiv_pass<0>(sdv, so, d1[0], m16, half, Bc);
        graddiv_pass<1>(sdv, so, d1[1], m16, half, Bc);
        graddiv_pass<2>(sdv, so, d1[2], m16, half, Bc);

        // ---- async drain result tile: LDS -> HBM, 12 x b128 per lane ----
        wait_ds0();         // ASYNCcnt is unordered vs DScnt: out accum must be in LDS
        unsigned so_lds = (unsigned)(uintptr_t)so;
        unsigned long long go = (unsigned long long)(uintptr_t)(outg + (size_t)batch * NFIELD);
#pragma unroll
        for (int i = 0; i < 12; ++i) {
            unsigned off = (unsigned)(lane + i * 32) * 16u;
            async_store_b128(go + off, so_lds + off);
        }
        wait_async0();
    }
}

extern "C" void kernel_launch(void* const* d_in, const int* in_sizes, int n_in,
                              void* d_out, int out_size, void* d_ws, size_t ws_size,
                              hipStream_t stream) {
    const float* state = (const float*)d_in[0];
    const float* D1    = (const float*)d_in[1];
    const float* D2    = (const float*)d_in[2];
    const float* p     = (const float*)d_in[3];
    float* out         = (float*)d_out;

    const int nbatch = in_sizes[0] / NFIELD;
    const int wgs = (nbatch + WAVES_PER_WG - 1) / WAVES_PER_WG;
    const size_t smem_bytes = (size_t)SMEM_FLOATS * sizeof(float);

    (void)hipFuncSetAttribute((const void*)fluid_rhs_kernel,
                              hipFuncAttributeMaxDynamicSharedMemorySize,
                              (int)smem_bytes);

    fluid_rhs_kernel<<<wgs, WAVES_PER_WG * 32, smem_bytes, stream>>>(
        state, D1, D2, p, out, nbatch);
}